// Agg_666_23021024706996
// MI455X (gfx1250) — compile-verified
//
#include <hip/hip_runtime.h>
#include <hip/hip_bf16.h>

// ---------------------------------------------------------------------------
// Problem constants (from reference setup_inputs)
// ---------------------------------------------------------------------------
#define BB   16          // batch
#define NN   16          // initial slots
#define CC   512         // channels
#define HH   8
#define WW   8
#define MM   31          // total slots (2n-1)
#define MP   32          // padded slots for WMMA tiles
#define CHW  32768       // C*H*W features per slot
#define KC   128         // K-chunk staged in LDS
#define KCP  132         // padded LDS row stride (floats): 132%64=4 -> conflict free, 528B row (16B aligned)

typedef __attribute__((ext_vector_type(2))) float v2f;
typedef __attribute__((ext_vector_type(8))) float v8f;

// ---------------------------------------------------------------------------
// init: xb[b][s] = x[b][s] for s<16, zeros for s in [16,32); mask init
// ---------------------------------------------------------------------------
__global__ void init_xb_kernel(const float* __restrict__ x, float* __restrict__ xb) {
    int blk = blockIdx.x;                 // 0 .. B*MP-1
    int b = blk >> 5;
    int s = blk & 31;
    float* dst = xb + ((size_t)b * MP + s) * CHW;
    if (s < NN) {
        const float* src = x + ((size_t)b * NN + s) * CHW;
        for (int e = threadIdx.x; e < CHW; e += 256) dst[e] = src[e];
    } else {
        for (int e = threadIdx.x; e < CHW; e += 256) dst[e] = 0.0f;
    }
}

__global__ void init_mask_kernel(float* __restrict__ mask) {
    int t = threadIdx.x;                  // 512 threads = B*32
    int s = t & 31;
    mask[t] = (s < NN) ? 0.0f : __builtin_inff();
}

// ---------------------------------------------------------------------------
// Gram: per batch, G(32x32) = T * T^T over K=32768, via V_WMMA_F32_16X16X4_F32.
// Block = 128 threads (4 waves); wave w owns 16x16 tile (w>>1, w&1).
// K staged through LDS in 128-wide chunks.
// ---------------------------------------------------------------------------
__global__ void gram_kernel(const float* __restrict__ xb, float* __restrict__ gram) {
    __shared__ __align__(16) float lds[MP * KCP];

    const int b    = blockIdx.x;
    const int t    = threadIdx.x;         // 0..127
    const int wave = t >> 5;
    const int lane = t & 31;
    const int ti   = (wave >> 1) * 16;    // tile row base (slot)
    const int tj   = (wave & 1) * 16;     // tile col base (slot)
    const int hv   = lane >> 4;           // 0: K=k+0..1  1: K=k+2..3
    const int lm   = lane & 15;

    const float* base = xb + (size_t)b * MP * CHW;

    // loader mapping: 4 threads per slot, 8 float4 each (32 float4 per 128-f row)
    const int lslot = t >> 2;
    const int lc4   = t & 3;

    v8f acc = {};

    for (int k0 = 0; k0 < CHW; k0 += KC) {
        __syncthreads();                  // previous chunk fully consumed
        {
            const float* gsrc = base + (size_t)lslot * CHW + k0;
            float*       ldst = &lds[lslot * KCP];
#pragma unroll
            for (int q = 0; q < 8; ++q) {
                int f = (lc4 + q * 4) * 4;
                *(float4*)&ldst[f] = *(const float4*)&gsrc[f];
            }
        }
        __syncthreads();

#pragma unroll 4
        for (int kk = 0; kk < KC; kk += 4) {
            const int kb = kk + 2 * hv;
            const float* pa = &lds[(ti + lm) * KCP + kb];   // A: row = slot ti+lm
            const float* pb = &lds[(tj + lm) * KCP + kb];   // B: col = slot tj+lm (T^T)
            v2f a;  a.x  = pa[0]; a.y  = pa[1];
            v2f bf; bf.x = pb[0]; bf.y = pb[1];
            acc = __builtin_amdgcn_wmma_f32_16x16x4_f32(
                /*neg_a=*/false, a, /*neg_b=*/false, bf,
                /*c_mod=*/(short)0, acc, /*reuse_a=*/false, /*reuse_b=*/false);
        }
    }

    // C/D layout: VGPR r, lanes 0-15 -> M=r, N=lane; lanes 16-31 -> M=r+8
    float* gout = gram + b * 1024;
#pragma unroll
    for (int r = 0; r < 8; ++r) {
        gout[(ti + r + 8 * hv) * 32 + (tj + lm)] = acc[r];
    }
}

// ---------------------------------------------------------------------------
// Argmin over masked pairwise distances; updates mask; records (i,j).
// Tie-break = smallest flat index i*31+j (matches jnp.argmin on flattened).
// ---------------------------------------------------------------------------
__global__ void argmin_kernel(const float* __restrict__ gram,
                              float* __restrict__ mask,
                              int* __restrict__ pairIdx, int slot) {
    __shared__ float sval[256];
    __shared__ int   sidx[256];

    const int b = blockIdx.x;
    const int t = threadIdx.x;
    const float* g  = gram + b * 1024;
    const float* mk = mask + b * 32;

    float best = __builtin_inff();
    int   bi   = 0;
    for (int e = t; e < MM * MM; e += 256) {
        int i = e / MM;
        int j = e - i * MM;
        if (i != j) {
            float d = g[i * 33] + g[j * 33] - 2.0f * g[i * 32 + j] + mk[i] + mk[j];
            if (d < best) { best = d; bi = e; }   // strict '<' keeps earliest e
        }
    }
    sval[t] = best; sidx[t] = bi;
    __syncthreads();
    for (int s2 = 128; s2 > 0; s2 >>= 1) {
        if (t < s2) {
            float v = sval[t + s2]; int ix = sidx[t + s2];
            if (v < sval[t] || (v == sval[t] && ix < sidx[t])) { sval[t] = v; sidx[t] = ix; }
        }
        __syncthreads();
    }
    if (t == 0) {
        int e = sidx[0];
        int i = e / MM;
        int j = e - i * MM;
        pairIdx[b * 2 + 0] = i;
        pairIdx[b * 2 + 1] = j;
        mask[b * 32 + slot] = 0.0f;
        mask[b * 32 + i] = __builtin_inff();
        mask[b * 32 + j] = __builtin_inff();
    }
}

// ---------------------------------------------------------------------------
// Merge: per (batch, channel), 7x7 pad-3 cross-correlation of the (xl,xr)
// 2-channel stack with conv_w(1,2,7,7) + bias -> xb[:, slot].
// ---------------------------------------------------------------------------
__global__ void merge_kernel(float* __restrict__ xb,
                             const float* __restrict__ cw,
                             const float* __restrict__ cb,
                             const int* __restrict__ pairIdx, int slot) {
    __shared__ float tl[64], tr[64], wbuf[98];

    const int blk = blockIdx.x;           // 0 .. B*C-1
    const int b   = blk >> 9;             // /512
    const int ch  = blk & 511;
    const int t   = threadIdx.x;          // 0..63

    const int i = pairIdx[b * 2 + 0];
    const int j = pairIdx[b * 2 + 1];

    float* base = xb + (size_t)b * MP * CHW;
    tl[t] = base[(size_t)i * CHW + ch * 64 + t];
    tr[t] = base[(size_t)j * CHW + ch * 64 + t];
    if (t < 49) { wbuf[t] = cw[t]; wbuf[49 + t] = cw[49 + t]; }
    __syncthreads();

    const int y = t >> 3;
    const int x = t & 7;
    float a = cb[0];
#pragma unroll
    for (int dy = 0; dy < 7; ++dy) {
        int yy = y + dy - 3;
        if (yy < 0 || yy > 7) continue;
#pragma unroll
        for (int dx = 0; dx < 7; ++dx) {
            int xx = x + dx - 3;
            if (xx < 0 || xx > 7) continue;
            a += wbuf[dy * 7 + dx]      * tl[yy * 8 + xx]
               + wbuf[49 + dy * 7 + dx] * tr[yy * 8 + xx];
        }
    }
    base[(size_t)slot * CHW + ch * 64 + t] = a;
}

// ---------------------------------------------------------------------------
// Final: out = xb[:, 30]
// ---------------------------------------------------------------------------
__global__ void copy_out_kernel(const float* __restrict__ xb, float* __restrict__ out) {
    size_t idx = (size_t)blockIdx.x * 256 + threadIdx.x;   // 2048 blocks -> 524288
    int b = (int)(idx >> 15);            // / CHW
    int r = (int)(idx & (CHW - 1));
    out[idx] = xb[((size_t)b * MP + (MM - 1)) * CHW + r];
}

// ---------------------------------------------------------------------------
// Launch
// ---------------------------------------------------------------------------
extern "C" void kernel_launch(void* const* d_in, const int* in_sizes, int n_in,
                              void* d_out, int out_size, void* d_ws, size_t ws_size,
                              hipStream_t stream) {
    const float* x      = (const float*)d_in[0];   // (16,16,512,8,8)
    const float* conv_w = (const float*)d_in[1];   // (1,2,7,7)
    const float* conv_b = (const float*)d_in[2];   // (1,)
    float* out = (float*)d_out;

    // Workspace layout (floats)
    float* xb   = (float*)d_ws;                         // B*MP*CHW = 16M floats (64 MB)
    float* gram = xb + (size_t)BB * MP * CHW;           // B*1024
    float* mask = gram + (size_t)BB * 1024;             // B*32
    int*   pidx = (int*)(mask + (size_t)BB * 32);       // B*2

    (void)in_sizes; (void)n_in; (void)out_size; (void)ws_size;

    init_xb_kernel<<<BB * MP, 256, 0, stream>>>(x, xb);
    init_mask_kernel<<<1, BB * 32, 0, stream>>>(mask);

    for (int s = NN; s < MM; ++s) {
        gram_kernel<<<BB, 128, 0, stream>>>(xb, gram);
        argmin_kernel<<<BB, 256, 0, stream>>>(gram, mask, pidx, s);
        merge_kernel<<<BB * CC, 64, 0, stream>>>(xb, conv_w, conv_b, pidx, s);
    }

    copy_out_kernel<<<(BB * CHW) / 256, 256, 0, stream>>>(xb, out);
}